// GCNLayer_1305670058274
// MI455X (gfx1250) — compile-verified
//
#include <hip/hip_runtime.h>
#include <stdint.h>

// ---------------------------------------------------------------------------
// GCN layer for MI455X (gfx1250, wave32).
// Pipeline: degree -> norm/feats/nh -> edge cosine keys -> 4-pass radix select
// (top-k cut) -> edge weights (+stable tie-break) -> scatter aggregation ->
// f32 WMMA GEMM (V_WMMA_F32_16X16X4_F32) with fused norm/ReLU/residual.
// Everything is L2-resident on MI455X (192MB L2); GEMM is tiny (1.64 GFLOP)
// so full-precision f32 WMMA is the right precision choice.
// ---------------------------------------------------------------------------

typedef float v2f __attribute__((ext_vector_type(2)));
typedef float v8f __attribute__((ext_vector_type(8)));

#define N_NODESC 50000
#define N_EDGESC 800000
#define DDIM     128
#define K_CUT    80000          // int(800000 * 0.1)
#define M_PAD    50048          // ceil(50000/128)*128, zero-padded agg rows
#define EPS_NRM  1e-12f

// ---------------- init ----------------
__global__ void init_state_kernel(uint32_t* __restrict__ state,
                                  uint32_t* __restrict__ hist) {
    if (threadIdx.x == 0) { state[0] = 0u; state[1] = (uint32_t)K_CUT; }
    hist[threadIdx.x] = 0u;   // blockDim == 256
}

// ---------------- in-degree ----------------
__global__ void deg_kernel(const int* __restrict__ dst,
                           uint32_t* __restrict__ deg) {
    for (int e = blockIdx.x * blockDim.x + threadIdx.x; e < N_EDGESC;
         e += gridDim.x * blockDim.x)
        atomicAdd(&deg[dst[e]], 1u);
}

// ---------------- norm, scaled feats, unit feats ----------------
// one block (128 threads) per node
__global__ void norm_feats_kernel(const float* __restrict__ features,
                                  const uint32_t* __restrict__ deg,
                                  float* __restrict__ norm,
                                  float* __restrict__ feats,
                                  float* __restrict__ nh) {
    __shared__ float red[4];
    __shared__ float inv_s;
    const int node = blockIdx.x;
    const int d    = threadIdx.x;           // 0..127
    const int lane = d & 31, wv = d >> 5;

    const float nv = rsqrtf(fmaxf((float)deg[node], 1.0f));
    const float x  = features[(size_t)node * DDIM + d] * nv;
    feats[(size_t)node * DDIM + d] = x;

    float ss = x * x;
    ss += __shfl_xor(ss, 1);  ss += __shfl_xor(ss, 2);
    ss += __shfl_xor(ss, 4);  ss += __shfl_xor(ss, 8);
    ss += __shfl_xor(ss, 16);
    if (lane == 0) red[wv] = ss;
    __syncthreads();
    if (d == 0) {
        float tot = red[0] + red[1] + red[2] + red[3];
        inv_s = 1.0f / fmaxf(sqrtf(tot), EPS_NRM);
        norm[node] = nv;
    }
    __syncthreads();
    nh[(size_t)node * DDIM + d] = x * inv_s;
}

// ---------------- per-edge cosine -> orderable uint key ----------------
// 8 lanes per edge, 32 edges per 256-thread block
__global__ void cos_kernel(const float* __restrict__ nh,
                           const int* __restrict__ src,
                           const int* __restrict__ dst,
                           uint32_t* __restrict__ keys) {
    const int tid = threadIdx.x;
    const int e   = blockIdx.x * 32 + (tid >> 3);
    if (e >= N_EDGESC) return;
    const int sub = tid & 7;
    const float4* a = (const float4*)(nh + (size_t)src[e] * DDIM) + sub;
    const float4* b = (const float4*)(nh + (size_t)dst[e] * DDIM) + sub;
    float acc = 0.f;
#pragma unroll
    for (int j = 0; j < 4; ++j) {
        float4 av = a[j * 8], bv = b[j * 8];
        acc += av.x * bv.x + av.y * bv.y + av.z * bv.z + av.w * bv.w;
    }
    acc += __shfl_xor(acc, 1);
    acc += __shfl_xor(acc, 2);
    acc += __shfl_xor(acc, 4);
    if (sub == 0) {
        uint32_t u = __float_as_uint(acc);
        keys[e] = (u & 0x80000000u) ? ~u : (u | 0x80000000u); // ascending order
    }
}

// ---------------- radix select (4 byte passes, MSB first) ----------------
__global__ void hist_kernel(const uint32_t* __restrict__ keys,
                            uint32_t* __restrict__ hist,
                            const uint32_t* __restrict__ state, int pass) {
    __shared__ uint32_t h[256];
    h[threadIdx.x] = 0u;
    __syncthreads();
    const uint32_t prefix   = state[0];
    const uint32_t maskHigh = (pass == 3) ? 0u
                              : (0xFFFFFFFFu << ((pass + 1) * 8));
    for (int e = blockIdx.x * blockDim.x + threadIdx.x; e < N_EDGESC;
         e += gridDim.x * blockDim.x) {
        uint32_t k = keys[e];
        if ((k & maskHigh) == (prefix & maskHigh))
            atomicAdd(&h[(k >> (pass * 8)) & 255u], 1u);
    }
    __syncthreads();
    atomicAdd(&hist[threadIdx.x], h[threadIdx.x]);
}

__global__ void select_kernel(uint32_t* __restrict__ hist,
                              uint32_t* __restrict__ state, int pass) {
    __shared__ uint32_t sh[256];
    sh[threadIdx.x] = hist[threadIdx.x];
    __syncthreads();
    if (threadIdx.x == 0) {
        uint32_t kRem = state[1], cum = 0, b = 0;
        for (b = 0; b < 256; ++b) {
            if (cum + sh[b] >= kRem) break;
            cum += sh[b];
        }
        state[0] |= b << (pass * 8);
        state[1]  = kRem - cum;
    }
    __syncthreads();
    hist[threadIdx.x] = 0u;   // reset for next pass
}

// ---------------- edge weights: cut keys < T ----------------
__global__ void weight_kernel(const uint32_t* __restrict__ keys,
                              const uint32_t* __restrict__ state,
                              float* __restrict__ w) {
    const uint32_t T = state[0];
    for (int e = blockIdx.x * blockDim.x + threadIdx.x; e < N_EDGESC;
         e += gridDim.x * blockDim.x)
        w[e] = (keys[e] < T) ? 0.f : 1.f;
}

// ---------------- stable tie-break: cut first `need` edges with key==T ----
__global__ void tie_kernel(const uint32_t* __restrict__ keys,
                           const uint32_t* __restrict__ state,
                           float* __restrict__ w) {
    __shared__ uint32_t wsum[32];
    __shared__ uint32_t running_s;
    const uint32_t T    = state[0];
    const uint32_t need = state[1];
    if (need == 0u) return;
    const int tid = threadIdx.x, lane = tid & 31, wv = tid >> 5;
    if (tid == 0) running_s = 0u;
    __syncthreads();
    for (int base = 0; base < N_EDGESC; base += 1024) {
        const int e = base + tid;
        const bool pred = (e < N_EDGESC) && (keys[e] == T);
        const uint32_t bal = (uint32_t)__ballot(pred);
        const uint32_t pre = __popc(bal & ((1u << lane) - 1u));
        if (lane == 0) wsum[wv] = __popc(bal);
        __syncthreads();
        uint32_t waveBase = 0u, total = 0u;
        for (int i = 0; i < 32; ++i) {
            uint32_t c = wsum[i];
            if (i < wv) waveBase += c;
            total += c;
        }
        const uint32_t rank = running_s + waveBase + pre;
        if (pred && rank < need) w[e] = 0.f;
        __syncthreads();
        if (tid == 0) running_s += total;
        __syncthreads();
        if (running_s >= need) break;
    }
}

// ---------------- scatter aggregation: one wave per edge ----------------
__global__ void aggregate_kernel(const float* __restrict__ feats,
                                 const float* __restrict__ w,
                                 const int* __restrict__ src,
                                 const int* __restrict__ dst,
                                 float* __restrict__ agg) {
    const int e = blockIdx.x * 8 + (threadIdx.x >> 5);
    if (e >= N_EDGESC) return;
    if (w[e] == 0.f) return;
    const int lane = threadIdx.x & 31;
    const int s = src[e], d = dst[e];
    float4 v = *(const float4*)(feats + (size_t)s * DDIM + lane * 4);
    float* o = agg + (size_t)d * DDIM + lane * 4;
    atomicAdd(o + 0, v.x);
    atomicAdd(o + 1, v.y);
    atomicAdd(o + 2, v.z);
    atomicAdd(o + 3, v.w);
}

// ---------------- WMMA GEMM + fused epilogue ----------------
// out = relu((agg @ W^T) * norm) + features
// Block: 256 threads = 8 waves; each wave computes a 16x128 output strip via
// 8 accumulator tiles of V_WMMA_F32_16X16X4_F32 (K stepped by 4, A reused).
__global__ __launch_bounds__(256)
void gemm_kernel(const float* __restrict__ agg,      // [M_PAD][128], padded
                 const float* __restrict__ W,        // [128][128] row-major
                 const float* __restrict__ norm,     // [N]
                 const float* __restrict__ features, // [N][128]
                 float* __restrict__ out) {          // [N][128]
    __shared__ float sW[DDIM * DDIM];                // 64 KB, W staged in LDS
    const int tid = threadIdx.x;
    for (int i = tid; i < DDIM * DDIM / 4; i += 256)
        ((float4*)sW)[i] = ((const float4*)W)[i];
    __syncthreads();

    const int wv     = tid >> 5;
    const int lane   = tid & 31;
    const int lane16 = lane & 15;
    const int hi     = lane >> 4;                    // 0 or 1
    const int m0     = (blockIdx.x * 8 + wv) * 16;   // 16 node rows per wave

    v8f c[8];
#pragma unroll
    for (int t = 0; t < 8; ++t)
#pragma unroll
        for (int i = 0; i < 8; ++i) c[t][i] = 0.f;

    // A fragment: lane<16 -> A[m0+lane][k0..k0+1]; lane>=16 -> K=k0+2..k0+3
    const float* aBase = agg + (size_t)(m0 + lane16) * DDIM + 2 * hi;
    for (int k0 = 0; k0 < DDIM; k0 += 4) {
        v2f a = *(const v2f*)(aBase + k0);
#pragma unroll
        for (int t = 0; t < 8; ++t) {
            // B[k][n] = W[n][k]; lane holds column n = t*16+lane16
            v2f b = *(const v2f*)(sW + (t * 16 + lane16) * DDIM + k0 + 2 * hi);
            c[t] = __builtin_amdgcn_wmma_f32_16x16x4_f32(
                false, a, false, b, (short)0, c[t], false, false);
        }
    }

    // Epilogue: C VGPR v -> row m0 + 8*hi + v, col t*16 + lane16
#pragma unroll
    for (int v = 0; v < 8; ++v) {
        const int row = m0 + 8 * hi + v;
        if (row >= N_NODESC) continue;
        const float nv = norm[row];
        const size_t base = (size_t)row * DDIM + lane16;
#pragma unroll
        for (int t = 0; t < 8; ++t) {
            float h = fmaxf(c[t][v] * nv, 0.f);
            out[base + t * 16] = h + features[base + t * 16];
        }
    }
}

// ---------------------------------------------------------------------------
extern "C" void kernel_launch(void* const* d_in, const int* in_sizes, int n_in,
                              void* d_out, int out_size, void* d_ws,
                              size_t ws_size, hipStream_t stream) {
    const float* features = (const float*)d_in[0];
    const float* W        = (const float*)d_in[1];
    const int*   src      = (const int*)d_in[2];
    const int*   dst      = (const int*)d_in[3];
    float*       out      = (float*)d_out;
    (void)in_sizes; (void)n_in; (void)out_size; (void)ws_size;

    char* ws = (char*)d_ws;
    size_t off = 0;
    auto alloc = [&](size_t bytes) -> void* {
        void* p = ws + off;
        off = (off + bytes + 255) & ~(size_t)255;
        return p;
    };
    float*    norm  = (float*)alloc((size_t)N_NODESC * 4);
    uint32_t* deg   = (uint32_t*)alloc((size_t)N_NODESC * 4);
    float*    feats = (float*)alloc((size_t)N_NODESC * DDIM * 4);
    float*    nh    = (float*)alloc((size_t)N_NODESC * DDIM * 4);
    uint32_t* keys  = (uint32_t*)alloc((size_t)N_EDGESC * 4);
    float*    w     = (float*)alloc((size_t)N_EDGESC * 4);
    float*    agg   = (float*)alloc((size_t)M_PAD * DDIM * 4);
    uint32_t* hist  = (uint32_t*)alloc(256 * 4);
    uint32_t* state = (uint32_t*)alloc(64);

    hipMemsetAsync(deg, 0, (size_t)N_NODESC * 4, stream);
    hipMemsetAsync(agg, 0, (size_t)M_PAD * DDIM * 4, stream);
    init_state_kernel<<<1, 256, 0, stream>>>(state, hist);

    deg_kernel<<<1024, 256, 0, stream>>>(dst, deg);
    norm_feats_kernel<<<N_NODESC, 128, 0, stream>>>(features, deg, norm, feats, nh);
    cos_kernel<<<(N_EDGESC + 31) / 32, 256, 0, stream>>>(nh, src, dst, keys);
    for (int pass = 3; pass >= 0; --pass) {
        hist_kernel<<<256, 256, 0, stream>>>(keys, hist, state, pass);
        select_kernel<<<1, 256, 0, stream>>>(hist, state, pass);
    }
    weight_kernel<<<(N_EDGESC + 255) / 256, 256, 0, stream>>>(keys, state, w);
    tie_kernel<<<1, 1024, 0, stream>>>(keys, state, w);
    aggregate_kernel<<<(N_EDGESC + 7) / 8, 256, 0, stream>>>(feats, w, src, dst, agg);
    gemm_kernel<<<M_PAD / 128, 256, 0, stream>>>(agg, W, norm, features, out);
}